// kWTA_89696097009963
// MI455X (gfx1250) — compile-verified
//
#include <hip/hip_runtime.h>
#include <hip/hip_bf16.h>
#include <stdint.h>

// ---------------------------------------------------------------------------
// k-WTA: per-row exact radix-select of the 409th-largest value, threshold,
// ReLU, L1 normalize.  Memory-bound (~1 GB traffic -> ~44us floor @ 23.3TB/s).
// One workgroup (256 thr = 8 wave32) per row; row staged to LDS via the
// CDNA5 async global->LDS engine (ASYNCcnt).  Radix-select bin scan is done
// by a single wave32 in registers (shfl butterfly) -> only 3 barriers/pass.
// ---------------------------------------------------------------------------

#define KWTA_D     2048
#define KWTA_TPB   256
#define KWTA_EPT   8          // elements per thread
#define KWTA_RANK  408u       // 0-indexed descending rank of topv[:, k-2], k=410
#define KWTA_EPS   1e-8f

typedef int kwta_v4i __attribute__((ext_vector_type(4)));
typedef __attribute__((address_space(1))) kwta_v4i* kwta_gp;
typedef __attribute__((address_space(3))) kwta_v4i* kwta_lp;

// ---- CDNA5 async global->LDS copy (16B per lane) ---------------------------
__device__ __forceinline__ void kwta_async_b128(const void* g, void* l) {
#if __has_builtin(__builtin_amdgcn_global_load_async_to_lds_b128)
  __builtin_amdgcn_global_load_async_to_lds_b128(
      (kwta_gp)(uintptr_t)g,
      (kwta_lp)(uint32_t)(uintptr_t)l,
      0, 0);
#else
  unsigned lds_addr = (unsigned)(uintptr_t)l;
  unsigned long long gaddr = (unsigned long long)(uintptr_t)g;
  asm volatile("global_load_async_to_lds_b128 %0, %1, off"
               :: "v"(lds_addr), "v"(gaddr) : "memory");
#endif
}

__device__ __forceinline__ void kwta_wait_async0() {
#if __has_builtin(__builtin_amdgcn_s_wait_asynccnt)
  __builtin_amdgcn_s_wait_asynccnt(0);
#else
  asm volatile("s_wait_asynccnt 0" ::: "memory");
#endif
}

__global__ __launch_bounds__(KWTA_TPB)
void kwta_kernel(const float* __restrict__ x, float* __restrict__ out) {
  __shared__ __align__(16) float vals[KWTA_D];     // 8 KB row staging
  __shared__ unsigned hist[KWTA_TPB];              // 256-bin histogram
  __shared__ unsigned s_sel, s_t;
  __shared__ float    s_wsum[KWTA_TPB / 32];
  __shared__ float    s_sum;

  const int       tid  = threadIdx.x;
  const long long row  = blockIdx.x;
  const float* __restrict__ xrow = x   + row * (long long)KWTA_D;
  float*       __restrict__ orow = out + row * (long long)KWTA_D;

  // ---- stage row into LDS with async copies (2 x 4KB, 16B/lane) ----------
  {
    const char* gb = (const char*)xrow;
    char*       lb = (char*)vals;
#pragma unroll
    for (int c = 0; c < 2; ++c) {
      int off = c * 4096 + tid * 16;
      kwta_async_b128(gb + off, lb + off);
    }
  }
  kwta_wait_async0();
  __syncthreads();

  // ---- pull my 8 strided elements into registers; build orderable keys ----
  float    xv[KWTA_EPT];
  unsigned kv[KWTA_EPT];
  bool     act[KWTA_EPT];
#pragma unroll
  for (int i = 0; i < KWTA_EPT; ++i) {
    float v = vals[tid + i * KWTA_TPB];
    xv[i] = v;
    unsigned b = __float_as_uint(v);
    // monotone map: descending float order == descending uint order
    kv[i]  = (b & 0x80000000u) ? ~b : (b | 0x80000000u);
    act[i] = true;
  }

  // ---- 4-pass MSB-first radix select of descending rank KWTA_RANK ---------
  unsigned prefix = 0;
  unsigned t = KWTA_RANK;
#pragma unroll
  for (int pass = 0; pass < 4; ++pass) {
    const int shift = 24 - pass * 8;

    hist[tid] = 0;
    __syncthreads();
#pragma unroll
    for (int i = 0; i < KWTA_EPT; ++i) {
      if (act[i]) atomicAdd(&hist[(kv[i] >> shift) & 0xFFu], 1u);
    }
    __syncthreads();

    // ---- wave 0 only: suffix-scan of 256 bins in registers, zero barriers.
    // Lane L owns bins [8L, 8L+8); lsuf[j] = sum of h[j..7]; cross-lane
    // exclusive suffix via shfl_down butterfly.  suff(b) = #active >= bin b.
    if (tid < 32) {
      const int lane = tid;
      unsigned h[8];
#pragma unroll
      for (int j = 0; j < 8; ++j) h[j] = hist[lane * 8 + j];
      unsigned lsuf[9];
      lsuf[8] = 0u;
#pragma unroll
      for (int j = 7; j >= 0; --j) lsuf[j] = lsuf[j + 1] + h[j];
      const unsigned total = lsuf[0];
      unsigned v = total;                    // inclusive suffix over lanes
#pragma unroll
      for (int off = 1; off < 32; off <<= 1) {
        unsigned u = __shfl_down(v, off, 32);
        if (lane + off < 32) v += u;
      }
      const unsigned excl = v - total;       // sum over higher lanes
#pragma unroll
      for (int j = 0; j < 8; ++j) {
        unsigned suff  = lsuf[j]     + excl;
        unsigned suffn = lsuf[j + 1] + excl; // == suff(b+1), j==7 -> excl
        if (suff > t && suffn <= t) {        // unique (lane,j) hits window
          s_sel = (unsigned)(lane * 8 + j);
          s_t   = t - suffn;
        }
      }
    }
    __syncthreads();

    const unsigned sel = s_sel;
    t = s_t;
    prefix |= sel << shift;
#pragma unroll
    for (int i = 0; i < KWTA_EPT; ++i)
      act[i] = act[i] && (((kv[i] >> shift) & 0xFFu) == sel);
  }

  // ---- reconstruct threshold float from selected key ----------------------
  unsigned tbits = (prefix & 0x80000000u) ? (prefix ^ 0x80000000u) : ~prefix;
  const float thresh = __uint_as_float(tbits) + KWTA_EPS;

  // ---- ReLU(x - thresh), block-wide sum -----------------------------------
  float yv[KWTA_EPT];
  float lsum = 0.0f;
#pragma unroll
  for (int i = 0; i < KWTA_EPT; ++i) {
    float y = xv[i] - thresh;
    y = (y > 0.0f) ? y : 0.0f;
    yv[i] = y;
    lsum += y;
  }
#pragma unroll
  for (int off = 16; off > 0; off >>= 1)
    lsum += __shfl_xor(lsum, off, 32);          // wave32 butterfly reduce
  if ((tid & 31) == 0) s_wsum[tid >> 5] = lsum;
  __syncthreads();
  if (tid == 0) {
    float s = 0.0f;
#pragma unroll
    for (int w = 0; w < KWTA_TPB / 32; ++w) s += s_wsum[w];
    s_sum = s + KWTA_EPS;
  }
  __syncthreads();

  const float inv = 1.0f / s_sum;
#pragma unroll
  for (int i = 0; i < KWTA_EPT; ++i) {
    // streamed output: non-temporal store, keep L2 for upcoming input rows
    __builtin_nontemporal_store(yv[i] * inv, &orow[tid + i * KWTA_TPB]);
  }
}

extern "C" void kernel_launch(void* const* d_in, const int* in_sizes, int n_in,
                              void* d_out, int out_size, void* d_ws, size_t ws_size,
                              hipStream_t stream) {
  (void)n_in; (void)out_size; (void)d_ws; (void)ws_size;
  const float* x = (const float*)d_in[0];
  float* out = (float*)d_out;
  const int rows = in_sizes[0] / KWTA_D;   // 65536
  kwta_kernel<<<dim3(rows), dim3(KWTA_TPB), 0, stream>>>(x, out);
}